// ModelNew_66657892434195
// MI455X (gfx1250) — compile-verified
//
#include <hip/hip_runtime.h>

typedef __attribute__((ext_vector_type(2))) float v2f;
typedef __attribute__((ext_vector_type(8))) float v8f;

#define N_       32
#define C_       384
#define H_       56
#define W_       56
#define HW_      (H_ * W_)            // 3136
#define POS_TILE 64
#define TILES_PER_IMG (HW_ / POS_TILE) // 49 (exact)
#define KCHUNK   128
#define NCHUNK   (C_ / KCHUNK)        // 3
#define LDS_STRIDE 72                 // 128*72*4 = 36864 B; +8 dwords/row => half-wave A reads hit disjoint bank sets

__global__ __launch_bounds__(256) void dwpw_fused_wmma_f32(
    const float* __restrict__ x,     // (32,384,56,56)
    const float* __restrict__ w_dw,  // (384,1,3,3)
    const float* __restrict__ w_pw,  // (384,384,1,1)
    float* __restrict__ y)           // (32,384,56,56)
{
    __shared__ float dwbuf[KCHUNK * LDS_STRIDE];

    const int tid  = threadIdx.x;
    const int lane = tid & 31;
    const int wave = tid >> 5;
    const int ml   = lane & 15;       // column / M-row selector within half-wave
    const int lh   = lane >> 4;       // half-wave: selects K=0/1 vs K=2/3 (A,B), M or M+8 (C/D)

    const int wg   = blockIdx.x;
    const int n    = wg / TILES_PER_IMG;
    const int pos0 = (wg % TILES_PER_IMG) * POS_TILE;

    const int mb   = (wave & 3) * 16;     // 16-position sub-tile per wave
    const int ocg  = (wave >> 2) * 192;   // waves 0-3: oc 0..191, waves 4-7: oc 192..383

    v8f acc[12] = {};                     // 12 oc-tiles of 16 -> 96 accumulator VGPRs

    for (int kc = 0; kc < NCHUNK; ++kc) {
        if (kc) __syncthreads();          // previous chunk's GEMM finished reading LDS

        // ---------- phase 1: depthwise 3x3 into LDS (128 channels x 64 positions) ----------
        #pragma unroll 4
        for (int i = 0; i < (KCHUNK * POS_TILE) / 256; ++i) {   // 32 iterations
            const int idx = i * 256 + tid;
            const int cl  = idx >> 6;        // local channel 0..127
            const int pos = idx & 63;        // local position 0..63 (coalesced x reads)
            const int c   = kc * KCHUNK + cl;
            const int p   = pos0 + pos;
            const int h   = p / W_;
            const int w   = p % W_;
            const float* xp = x + ((n * C_ + c) * H_ + h) * W_ + w;
            const float* wd = w_dw + c * 9;
            float s = 0.f;
            #pragma unroll
            for (int kh = 0; kh < 3; ++kh) {
                const int hh = h + kh - 1;
                if (hh < 0 || hh >= H_) continue;
                #pragma unroll
                for (int kw = 0; kw < 3; ++kw) {
                    const int ww = w + kw - 1;
                    if (ww < 0 || ww >= W_) continue;
                    s += xp[(kh - 1) * W_ + (kw - 1)] * wd[kh * 3 + kw];
                }
            }
            dwbuf[cl * LDS_STRIDE + pos] = s;
        }
        __syncthreads();

        // ---------- phase 2: pointwise GEMM via V_WMMA_F32_16X16X4_F32 ----------
        // A (16x4): M=pos, K=c.  VGPR0: K=0 (lanes 0-15) / K=2 (lanes 16-31); VGPR1: K=1 / K=3.
        // B (4x16): K=c, N=oc.   Same K split by half-wave; N = lane&15.
        const int k0 = lh * 2;
        #pragma unroll 4
        for (int kk = 0; kk < KCHUNK / 4; ++kk) {   // 32 K-steps of 4
            const int cb = kk * 4;
            v2f a;
            a.x = dwbuf[(cb + k0    ) * LDS_STRIDE + mb + ml];
            a.y = dwbuf[(cb + k0 + 1) * LDS_STRIDE + mb + ml];
            #pragma unroll
            for (int t = 0; t < 12; ++t) {
                const int oc = ocg + t * 16 + ml;
                const float* wp = w_pw + oc * C_ + kc * KCHUNK + cb + k0; // 8B-aligned
                const v2f b = *(const v2f*)wp;                           // {w[oc][c+k0], w[oc][c+k0+1]}
                acc[t] = __builtin_amdgcn_wmma_f32_16x16x4_f32(
                    /*neg_a=*/false, a, /*neg_b=*/false, b,
                    /*c_mod=*/(short)0, acc[t],
                    /*reuse_a=*/false, /*reuse_b=*/false);
            }
        }
    }

    // ---------- phase 3: store D tiles ----------
    // D layout: VGPR v holds row M = v + 8*lh, column N = lane&15.
    // Per lane the 8 values are 8 consecutive spatial positions -> contiguous 32B.
    #pragma unroll
    for (int t = 0; t < 12; ++t) {
        const int oc = ocg + t * 16 + ml;
        float* yp = y + (n * C_ + oc) * HW_ + pos0 + mb + lh * 8;
        #pragma unroll
        for (int v = 0; v < 8; ++v) {
            yp[v] = acc[t][v];
        }
    }
}

extern "C" void kernel_launch(void* const* d_in, const int* in_sizes, int n_in,
                              void* d_out, int out_size, void* d_ws, size_t ws_size,
                              hipStream_t stream) {
    (void)in_sizes; (void)n_in; (void)out_size; (void)d_ws; (void)ws_size;
    const float* x    = (const float*)d_in[0];
    const float* w_dw = (const float*)d_in[1];
    const float* w_pw = (const float*)d_in[2];
    float* yout       = (float*)d_out;

    dim3 grid(N_ * TILES_PER_IMG);   // 1568 workgroups
    dim3 block(256);                 // 8 waves (wave32)
    dwpw_fused_wmma_f32<<<grid, block, 0, stream>>>(x, w_dw, w_pw, yout);
}